// MultiEmbedding_1726576854660
// MI455X (gfx1250) — compile-verified
//
#include <hip/hip_runtime.h>
#include <hip/hip_bf16.h>
#include <stdint.h>

// MultiEmbedding gather-sum for MI455X (gfx1250).
//
// out1[n, d] = sum_{l=0..7} weight[l, x1[n,l], d]   (n < 8192, d < 1024)
// out2[n, d] = sum_{l=0..3} weight[l, x2[n,l], d]
//
// Roofline: 32 MB weight table fits in the 192 MB L2; limiter is L2 gather
// bandwidth (384 MB of L2-resident reads) + 64 MB streamed output (~4 us of
// HBM at 23.3 TB/s). Arithmetic intensity ~0.2 FLOP/B -> WMMA inapplicable;
// the win is wide b128 loads, scalar-path index fetch, deep load pipelining.

#define TOKEN_DIM 1024
#define N_TOKENS  1024

// Native vector type: legal for __builtin_nontemporal_store, lowers to b128.
typedef float v4f __attribute__((ext_vector_type(4)));

template <int L>
__device__ __forceinline__ void embed_row(const int* __restrict__ x,
                                          const float* __restrict__ w,
                                          float* __restrict__ out, int row) {
  const int t = threadIdx.x;  // 0..255 -> one float4 of the 1024-wide row

  // Block-uniform index loads -> scalar memory path (s_load_bXXX).
  // All address math in unsigned 32-bit: per-level byte offset
  // ((l*1024 + idx) * 1024 floats) * 4 B < 2^25, so no sign-extension chains.
  const char* wbytes = reinterpret_cast<const char*>(w);
  const v4f* srcs[L];
#pragma unroll
  for (int l = 0; l < L; ++l) {
    const uint32_t idx = (uint32_t)x[row * L + l];
    const uint32_t byte_off = (((uint32_t)l << 10) + idx) << 12;  // *4096 B/row
    srcs[l] = reinterpret_cast<const v4f*>(wbytes + byte_off);
  }

  // Issue all L global_load_b128 before accumulating (independent addresses,
  // up to 128 B in flight per lane; LOADcnt pipelining hides L2 latency).
  v4f v[L];
#pragma unroll
  for (int l = 0; l < L; ++l) {
    v[l] = srcs[l][t];
  }

  v4f acc = v[0];
#pragma unroll
  for (int l = 1; l < L; ++l) {
    acc += v[l];
  }

  // Write-once output: non-temporal store so the result stream does not
  // rinse the L2-resident weight table.
  v4f* dst = reinterpret_cast<v4f*>(out + (size_t)row * TOKEN_DIM);
  __builtin_nontemporal_store(acc, dst + t);
}

__global__ __launch_bounds__(256, 8) void multi_embed_kernel(
    const int* __restrict__ x1,   // [n1, 8]
    const int* __restrict__ x2,   // [n2, 4]
    const float* __restrict__ w,  // [8, 1024, 1024]
    float* __restrict__ out,      // [n1 + n2, 1024] (out1 then out2)
    int n1, int n2) {
  const int b = blockIdx.x;  // block-uniform branch: [0,n1)->L=8, rest->L=4
  if (b < n1) {
    embed_row<8>(x1, w, out, b);
  } else if (b < n1 + n2) {
    embed_row<4>(x2, w, out + (size_t)n1 * TOKEN_DIM, b - n1);
  }
}

extern "C" void kernel_launch(void* const* d_in, const int* in_sizes, int n_in,
                              void* d_out, int out_size, void* d_ws,
                              size_t ws_size, hipStream_t stream) {
  // Inputs in setup_inputs() order: x1 [8192,8] i32, x2 [8192,4] i32,
  // weight [8,1024,1024] f32. Output: out1 then out2, flat f32.
  const int* x1 = (const int*)d_in[0];
  const int* x2 = (const int*)d_in[1];
  const float* w = (const float*)d_in[2];
  float* out = (float*)d_out;

  const int n1 = in_sizes[0] / 8;  // 8192
  const int n2 = in_sizes[1] / 4;  // 8192

  multi_embed_kernel<<<n1 + n2, 256, 0, stream>>>(x1, x2, w, out, n1, n2);
}